// GINEConvMultiEdgeset_13666585935969
// MI455X (gfx1250) — compile-verified
//
#include <hip/hip_runtime.h>
#include <hip/hip_bf16.h>
#include <math.h>

typedef float v2f __attribute__((ext_vector_type(2)));
typedef float v8f __attribute__((ext_vector_type(8)));

#define D_FEAT 64
#define H_FEAT 128
#define WAVES_PER_BLOCK 4

// ---------------------------------------------------------------------------
// Phase 1: edge messages. One wave32 per edge, handling BOTH channels so the
// bond-embedding row and edge indices are fetched once. Each lane owns two
// consecutive feature elements (float2 loads -> exact gelu -> f32 atomics).
// agg fits in L2 (25.6 MB << 192 MB), so atomics resolve on-chip.
// ---------------------------------------------------------------------------
__global__ void edge_msg_kernel(const float* __restrict__ x,
                                const int*   __restrict__ edge_index,
                                const int*   __restrict__ edge_attr,
                                const float* __restrict__ edge_weight,
                                const float* __restrict__ bond_emb,
                                float*       __restrict__ agg,
                                int E, int N)
{
    int e    = (int)((blockIdx.x * blockDim.x + threadIdx.x) >> 5);
    int lane = threadIdx.x & 31;
    if (e >= E) return;

    int src = edge_index[e];
    int dst = edge_index[E + e];
    int att = edge_attr[e];
    float w0 = edge_weight[e];              // channel 0
    float w1 = edge_weight[(size_t)E + e];  // channel 1

    int d = lane * 2;
    const float* er = bond_emb + (size_t)att * D_FEAT;
    float2 ev = *(const float2*)(er + d);

    const float kInvSqrt2 = 0.70710678118654752f;

    // channel 0
    {
        const float* xr = x   + ((size_t)src) * D_FEAT;
        float*       ar = agg + ((size_t)dst) * D_FEAT;
        float2 xv = *(const float2*)(xr + d);
        float v0 = xv.x + ev.x;
        float v1 = xv.y + ev.y;
        v0 = 0.5f * v0 * (1.0f + erff(v0 * kInvSqrt2));
        v1 = 0.5f * v1 * (1.0f + erff(v1 * kInvSqrt2));
        atomicAdd(ar + d,     v0 * w0);
        atomicAdd(ar + d + 1, v1 * w0);
    }
    // channel 1
    {
        const float* xr = x   + ((size_t)N + src) * D_FEAT;
        float*       ar = agg + ((size_t)N + dst) * D_FEAT;
        float2 xv = *(const float2*)(xr + d);
        float v0 = xv.x + ev.x;
        float v1 = xv.y + ev.y;
        v0 = 0.5f * v0 * (1.0f + erff(v0 * kInvSqrt2));
        v1 = 0.5f * v1 * (1.0f + erff(v1 * kInvSqrt2));
        atomicAdd(ar + d,     v0 * w1);
        atomicAdd(ar + d + 1, v1 * w1);
    }
}

// ---------------------------------------------------------------------------
// Phase 2: fused GIN MLP with fp32 WMMA (V_WMMA_F32_16X16X4_F32).
// One wave per 16-row tile of h = (1+eps)*x + agg.
//   t   = relu(h @ W1 + b1)   : 8 N-tiles x 16 K-steps of WMMA
//   out = t @ W2 + b2         : 4 N-tiles x 32 K-steps of WMMA
// A-layout (16x4 f32): lane m = L%16 holds row m; lane-half selects K pair.
// C/D-layout (16x16 f32): vgpr v, lane L -> row v+8*(L/16), col L%16.
// ---------------------------------------------------------------------------
__global__ void gin_mlp_kernel(const float* __restrict__ x,
                               const float* __restrict__ agg,
                               const float* __restrict__ W1,
                               const float* __restrict__ b1,
                               const float* __restrict__ W2,
                               const float* __restrict__ b2,
                               const float* __restrict__ eps,
                               float*       __restrict__ out,
                               int n_tiles)
{
    __shared__ float lds_t[WAVES_PER_BLOCK][16 * H_FEAT];

    int waveId = threadIdx.x >> 5;
    int lane   = threadIdx.x & 31;
    int tile   = blockIdx.x * WAVES_PER_BLOCK + waveId;
    if (tile >= n_tiles) return;   // whole-wave exit: EXEC stays all-1 for WMMA

    float epsv = 1.0f + eps[0];
    int m    = lane & 15;   // row within tile (A) / column (B,C,D)
    int half = lane >> 4;   // K-pair selector

    size_t rowbase = ((size_t)tile * 16 + m) * D_FEAT;

    // Build all 16 A-tiles (K=64) of h in WMMA A-layout.
    v2f A[16];
#pragma unroll
    for (int k = 0; k < 16; ++k) {
        int col = k * 4 + half * 2;
        float2 xv = *(const float2*)(x   + rowbase + col);
        float2 av = *(const float2*)(agg + rowbase + col);
        A[k].x = epsv * xv.x + av.x;
        A[k].y = epsv * xv.y + av.y;
    }

    float* tl = &lds_t[waveId][0];

    // ---- GEMM 1: t = relu(h @ W1 + b1), 16x128 result -> LDS ----
#pragma unroll
    for (int n = 0; n < 8; ++n) {
        v8f acc = {};
        int ncol = n * 16 + m;
#pragma unroll
        for (int k = 0; k < 16; ++k) {
            int kb = k * 4 + half * 2;
            v2f B;
            B.x = W1[(size_t)kb * H_FEAT + ncol];
            B.y = W1[(size_t)(kb + 1) * H_FEAT + ncol];
            acc = __builtin_amdgcn_wmma_f32_16x16x4_f32(
                      false, A[k], false, B, (short)0, acc, false, false);
        }
        float bias = b1[ncol];
#pragma unroll
        for (int v = 0; v < 8; ++v) {
            float t = acc[v] + bias;
            t = t > 0.0f ? t : 0.0f;
            tl[(v + 8 * half) * H_FEAT + ncol] = t;   // row-major 16x128 in LDS
        }
    }

    // wave-local LDS ordering: drain DS ops, keep scheduler from hoisting loads
    asm volatile("s_wait_dscnt 0" ::: "memory");
    __builtin_amdgcn_wave_barrier();

    // ---- GEMM 2: out = t @ W2 + b2, 16x64 result ----
#pragma unroll
    for (int o = 0; o < 4; ++o) {
        v8f acc = {};
        int ocol = o * 16 + m;
#pragma unroll
        for (int j = 0; j < 32; ++j) {
            int kb = j * 4 + half * 2;
            float2 av = *(const float2*)(tl + m * H_FEAT + kb);
            v2f A2; A2.x = av.x; A2.y = av.y;
            v2f B2;
            B2.x = W2[(size_t)kb * D_FEAT + ocol];
            B2.y = W2[(size_t)(kb + 1) * D_FEAT + ocol];
            acc = __builtin_amdgcn_wmma_f32_16x16x4_f32(
                      false, A2, false, B2, (short)0, acc, false, false);
        }
        float bias = b2[ocol];
#pragma unroll
        for (int v = 0; v < 8; ++v) {
            out[((size_t)tile * 16 + v + 8 * half) * D_FEAT + ocol] = acc[v] + bias;
        }
    }
}

extern "C" void kernel_launch(void* const* d_in, const int* in_sizes, int n_in,
                              void* d_out, int out_size, void* d_ws, size_t ws_size,
                              hipStream_t stream)
{
    const float* x           = (const float*)d_in[0];
    const int*   edge_index  = (const int*)  d_in[1];
    const int*   edge_attr   = (const int*)  d_in[2];
    const float* edge_weight = (const float*)d_in[3];
    const float* bond_emb    = (const float*)d_in[4];
    const float* eps         = (const float*)d_in[5];
    const float* W1          = (const float*)d_in[6];
    const float* b1          = (const float*)d_in[7];
    const float* W2          = (const float*)d_in[8];
    const float* b2          = (const float*)d_in[9];
    float*       out         = (float*)d_out;

    const int RC = 2;                           // R*C
    const int E  = in_sizes[2];                 // edge_attr count = 800000
    const int N  = in_sizes[0] / (RC * D_FEAT); // 50000

    // workspace: aggregation buffer (RC * N * D) — zero it every call
    float* agg = (float*)d_ws;
    hipMemsetAsync(agg, 0, (size_t)RC * N * D_FEAT * sizeof(float), stream);

    // phase 1: one wave per edge (both channels in-wave)
    long long totalThreads = (long long)E * 32;
    int blocks1 = (int)((totalThreads + 255) / 256);
    edge_msg_kernel<<<blocks1, 256, 0, stream>>>(x, edge_index, edge_attr,
                                                 edge_weight, bond_emb, agg, E, N);

    // phase 2: WMMA MLP, one wave per 16-row tile
    int n_tiles = (RC * N) / 16;                // 6250 (exact)
    int blocks2 = (n_tiles + WAVES_PER_BLOCK - 1) / WAVES_PER_BLOCK;
    gin_mlp_kernel<<<blocks2, WAVES_PER_BLOCK * 32, 0, stream>>>(
        x, agg, W1, b1, W2, b2, eps, out, n_tiles);
}